// edGNNLayer_64158221468060
// MI455X (gfx1250) — compile-verified
//
#include <hip/hip_runtime.h>

typedef __attribute__((ext_vector_type(2))) float v2f;
typedef __attribute__((ext_vector_type(4))) float v4f;
typedef __attribute__((ext_vector_type(8))) float v8f;

#define NODE_DIM 64
#define OUT_F    64
#define KDIM     192   // 2*NODE_DIM + EDGE_DIM
#define WPAD     65    // padded LDS stride for W
#define APAD     196   // padded LDS stride for A block
#define ROWS_PER_BLOCK 64

// ---------------- zero the msg accumulators in workspace ----------------
__global__ __launch_bounds__(256) void zero_ws_kernel(float* __restrict__ p, int n4) {
  int i = blockIdx.x * blockDim.x + threadIdx.x;
  if (i < n4) ((float4*)p)[i] = make_float4(0.f, 0.f, 0.f, 0.f);
}

// ---------------- per-edge scatter-add (segment sum) ----------------
// One wave32 per edge: lane L handles columns 2L, 2L+1. src/dst are
// wave-uniform -> scalarize. edge_features streams once -> non-temporal so
// the L2-resident working set (nf + the two msg accumulators, ~77 MB of the
// 192 MB L2) is not evicted.
__global__ __launch_bounds__(256) void scatter_edges_kernel(
    const float* __restrict__ nf, const float* __restrict__ ef,
    const int* __restrict__ src, const int* __restrict__ dst,
    float* __restrict__ msg_h, float* __restrict__ msg_e, int n_edges) {
  unsigned tid = blockIdx.x * blockDim.x + threadIdx.x;
  unsigned e = tid >> 5;
  if (e >= (unsigned)n_edges) return;
  unsigned c = (tid & 31u) * 2u;

  int s = __builtin_amdgcn_readfirstlane(src[e]);
  int d = __builtin_amdgcn_readfirstlane(dst[e]);

  const v2f hv = *(const v2f*)(nf + (size_t)s * NODE_DIM + c);
  const v2f ev = __builtin_nontemporal_load((const v2f*)(ef + (size_t)e * NODE_DIM + c));

  float* ph = msg_h + (size_t)d * NODE_DIM + c;
  float* pe = msg_e + (size_t)d * NODE_DIM + c;
  __hip_atomic_fetch_add(ph + 0, hv[0], __ATOMIC_RELAXED, __HIP_MEMORY_SCOPE_AGENT);
  __hip_atomic_fetch_add(ph + 1, hv[1], __ATOMIC_RELAXED, __HIP_MEMORY_SCOPE_AGENT);
  __hip_atomic_fetch_add(pe + 0, ev[0], __ATOMIC_RELAXED, __HIP_MEMORY_SCOPE_AGENT);
  __hip_atomic_fetch_add(pe + 1, ev[1], __ATOMIC_RELAXED, __HIP_MEMORY_SCOPE_AGENT);
}

// ---------------- fused concat + GEMM + bias + ReLU via f32 WMMA ----------------
// Block = 128 threads (4 waves) handles 64 node rows as 4 slabs of 16,
// reusing the LDS-staged W (amortizes W L2->LDS traffic 4x). Per slab each
// wave computes one 16x16 tile with 48 V_WMMA_F32_16X16X4_F32 ops.
__global__ __launch_bounds__(128) void gnn_gemm_kernel(
    const float* __restrict__ nf, const float* __restrict__ msg_h,
    const float* __restrict__ msg_e, const float* __restrict__ W,
    const float* __restrict__ bias, float* __restrict__ out, int n_nodes) {
  __shared__ float Ws[KDIM][WPAD];  // 192 x 65 f32 = 49,920 B
  __shared__ float As[16][APAD];    //  16 x 196 f32 = 12,544 B

  const int tid = threadIdx.x;

  // Stage W into LDS with float4 global loads (scalar LDS stores: WPAD
  // padding breaks 16B alignment on odd rows).
  for (int i = tid; i < (KDIM * OUT_F) / 4; i += 128) {
    int row = i >> 4;              // 16 float4 per 64-wide row
    int c4  = (i & 15) << 2;
    v4f v = *(const v4f*)(W + ((size_t)row << 6) + c4);
    Ws[row][c4 + 0] = v[0];
    Ws[row][c4 + 1] = v[1];
    Ws[row][c4 + 2] = v[2];
    Ws[row][c4 + 3] = v[3];
  }

  const int wave = tid >> 5;
  const int lane = tid & 31;
  const int half = lane >> 4;   // 0: lanes 0-15, 1: lanes 16-31
  const int mn   = lane & 15;   // row index for A, col index for B
  const int col0 = wave * 16;
  const float bv = bias[col0 + mn];

  for (int slab = 0; slab < ROWS_PER_BLOCK / 16; ++slab) {
    const int row0 = blockIdx.x * ROWS_PER_BLOCK + slab * 16;
    if (row0 >= n_nodes) break;          // block-uniform -> barrier-safe

    __syncthreads();  // As free of prior readers; also orders Ws staging (slab 0)

    // Stage the 16x192 concatenated A slab: [nf | msg_h | msg_e], float4-wide.
    for (int i = tid; i < 16 * (KDIM / 4); i += 128) {
      int r  = i / (KDIM / 4);
      int c4 = (i - r * (KDIM / 4)) * 4;
      int node = row0 + r;
      v4f v = {0.f, 0.f, 0.f, 0.f};
      if (node < n_nodes) {
        const float* sp = (c4 < 64)  ? nf    + ((size_t)node << 6) + c4
                        : (c4 < 128) ? msg_h + ((size_t)node << 6) + (c4 - 64)
                                     : msg_e + ((size_t)node << 6) + (c4 - 128);
        v = *(const v4f*)sp;
      }
      As[r][c4 + 0] = v[0];
      As[r][c4 + 1] = v[1];
      As[r][c4 + 2] = v[2];
      As[r][c4 + 3] = v[3];
    }
    __syncthreads();

    // A 16x4 f32 layout: lane L (M = L%16), VGPR j holds K = 2*(L/16) + j.
    // B  4x16 f32 layout: lane L (N = L%16), VGPR j holds K = 2*(L/16) + j.
    v8f acc = {};
    #pragma unroll 4
    for (int k = 0; k < KDIM; k += 4) {
      v2f a, b;
      a[0] = As[mn][k + 2 * half + 0];
      a[1] = As[mn][k + 2 * half + 1];
      b[0] = Ws[k + 2 * half + 0][col0 + mn];
      b[1] = Ws[k + 2 * half + 1][col0 + mn];
      acc = __builtin_amdgcn_wmma_f32_16x16x4_f32(
          /*neg_a=*/false, a, /*neg_b=*/false, b,
          /*c_mod=*/(short)0, acc, /*reuse_a=*/false, /*reuse_b=*/false);
    }

    // D layout: VGPR j, lane L -> M = j + 8*(L/16), N = L%16.
    // Output is written exactly once -> non-temporal stores.
    float* po = out + ((size_t)(row0 + 8 * half) << 6) + col0 + mn;
    if (row0 + 16 <= n_nodes) {          // fast path: no per-row exec masking
      #pragma unroll
      for (int j = 0; j < 8; ++j) {
        float v = acc[j] + bv;
        __builtin_nontemporal_store(v > 0.f ? v : 0.f, po + ((size_t)j << 6));
      }
    } else {
      #pragma unroll
      for (int j = 0; j < 8; ++j) {
        if (row0 + 8 * half + j < n_nodes) {
          float v = acc[j] + bv;
          __builtin_nontemporal_store(v > 0.f ? v : 0.f, po + ((size_t)j << 6));
        }
      }
    }
  }
}

extern "C" void kernel_launch(void* const* d_in, const int* in_sizes, int n_in,
                              void* d_out, int out_size, void* d_ws, size_t ws_size,
                              hipStream_t stream) {
  const float* nf   = (const float*)d_in[0];  // [N, 64]
  const float* ef   = (const float*)d_in[1];  // [E, 64]
  const int*   src  = (const int*)  d_in[2];  // [E]
  const int*   dst  = (const int*)  d_in[3];  // [E]
  const float* W    = (const float*)d_in[4];  // [192, 64]
  const float* bias = (const float*)d_in[5];  // [64]
  float*       out  = (float*)d_out;          // [N, 64]

  const int n_nodes = in_sizes[0] / NODE_DIM;
  const int n_edges = in_sizes[1] / NODE_DIM;

  float* msg_h = (float*)d_ws;                          // [N, 64]
  float* msg_e = msg_h + (size_t)n_nodes * NODE_DIM;    // [N, 64]

  // 1) zero the two accumulators (float4-granular)
  int n4 = (2 * n_nodes * NODE_DIM) / 4;
  zero_ws_kernel<<<(n4 + 255) / 256, 256, 0, stream>>>(msg_h, n4);

  // 2) per-edge scatter-add: one wave per edge
  long total_threads = (long)n_edges * 32;
  int  sblocks = (int)((total_threads + 255) / 256);
  scatter_edges_kernel<<<sblocks, 256, 0, stream>>>(nf, ef, src, dst, msg_h, msg_e, n_edges);

  // 3) fused concat+GEMM+bias+ReLU with f32 WMMA (64 rows per block)
  int gblocks = (n_nodes + ROWS_PER_BLOCK - 1) / ROWS_PER_BLOCK;
  gnn_gemm_kernel<<<gblocks, 128, 0, stream>>>(nf, msg_h, msg_e, W, bias, out, n_nodes);
}